// CorrectAndSmoothHeterophilyOGB_8108898254942
// MI455X (gfx1250) — compile-verified
//
#include <hip/hip_runtime.h>

// ---------------------------------------------------------------------------
// Correct & Smooth on MI455X (gfx1250).
// Pull-based CSR propagation: zero atomics in the 20 hot layers; entire
// ~47MB working set is L2-resident (192MB L2). CDNA5 async global->LDS
// staging (ASYNCcnt) on the CSR-build edge passes; smooth-phase softmax
// fused through LDS (blockDim=320 => the 10 threads of a node share a block).
// ---------------------------------------------------------------------------

static constexpr int KC = 40;  // number of classes (feature dim)

// ---- CDNA5 async global->LDS helpers (inline asm; ASYNCcnt tracked) -------
typedef __attribute__((address_space(3))) void lds_void_t;

__device__ __forceinline__ unsigned to_lds_off(void* p) {
  return (unsigned)(unsigned long long)(lds_void_t*)p;
}

__device__ __forceinline__ void async_load_b32(unsigned lds_byte_off, const void* gaddr) {
  asm volatile("global_load_async_to_lds_b32 %0, %1, off"
               :: "v"(lds_byte_off), "v"((unsigned long long)gaddr)
               : "memory");
}

__device__ __forceinline__ void wait_asynccnt0() {
  asm volatile("s_wait_asynccnt 0" ::: "memory");
}

// ---------------------------------------------------------------------------
__global__ void zero4_kernel(float4* __restrict__ p, int n4) {
  int i = blockIdx.x * blockDim.x + threadIdx.x;
  if (i < n4) p[i] = make_float4(0.f, 0.f, 0.f, 0.f);
}

// Per edge: deg[row] += w ; count[row] += 1. Edge stream staged via async->LDS.
__global__ void __launch_bounds__(256)
degcnt_kernel(const int* __restrict__ row, const float* __restrict__ w,
              float* __restrict__ deg, int* __restrict__ count, int E) {
  __shared__ int   s_row[256];
  __shared__ float s_w[256];
  const int t = threadIdx.x;
  const int e = blockIdx.x * 256 + t;
  if (e < E) {
    async_load_b32(to_lds_off(&s_row[t]), row + e);
    async_load_b32(to_lds_off(&s_w[t]),   w + e);
  }
  wait_asynccnt0();  // per-wave wait; each lane consumes only its own slot
  if (e < E) {
    const int r = s_row[t];
    atomicAdd(&deg[r], s_w[t]);
    atomicAdd(&count[r], 1);
  }
}

__global__ void rsqrt_kernel(float* __restrict__ deg, int N) {
  int i = blockIdx.x * blockDim.x + threadIdx.x;
  if (i < N) {
    float d = deg[i];
    deg[i] = (d > 0.f) ? rsqrtf(d) : 0.f;
  }
}

// Per node: carve a contiguous CSR segment via atomic cursor (no scan needed).
__global__ void alloc_kernel(const int* __restrict__ count, int2* __restrict__ seg,
                             int* __restrict__ wcur, int* __restrict__ cursor, int N) {
  int n = blockIdx.x * blockDim.x + threadIdx.x;
  if (n >= N) return;
  const int c = count[n];
  const int s = atomicAdd(cursor, c);
  seg[n]  = make_int2(s, c);
  wcur[n] = s;
}

// Per edge: drop (col, norm) into the row's CSR slot. Async->LDS staged.
__global__ void __launch_bounds__(256)
fill_kernel(const int* __restrict__ row, const int* __restrict__ col,
            const float* __restrict__ w, const float* __restrict__ dis,
            int* __restrict__ wcur, int2* __restrict__ epack, int E) {
  __shared__ int   s_row[256];
  __shared__ int   s_col[256];
  __shared__ float s_w[256];
  const int t = threadIdx.x;
  const int e = blockIdx.x * 256 + t;
  if (e < E) {
    async_load_b32(to_lds_off(&s_row[t]), row + e);
    async_load_b32(to_lds_off(&s_col[t]), col + e);
    async_load_b32(to_lds_off(&s_w[t]),   w + e);
  }
  wait_asynccnt0();
  if (e < E) {
    const int r = s_row[t];
    const int c = s_col[t];
    const float nrm = dis[r] * s_w[t] * dis[c];
    const int p = atomicAdd(&wcur[r], 1);
    epack[p] = make_int2(c, __float_as_int(nrm));
  }
}

// error = mask ? (one_hot - y_soft) : 0 ; accumulate sum|error| and count(mask)
__global__ void init_error_kernel(const int* __restrict__ yt, const float* __restrict__ ys,
                                  const unsigned char* __restrict__ mask,
                                  float* __restrict__ x, float* __restrict__ scal, int N) {
  int n = blockIdx.x * blockDim.x + threadIdx.x;
  if (n >= N) return;
  const bool m  = mask[n] != 0;
  const int lab = yt[n];
  const float4* ysr = (const float4*)(ys + (size_t)n * KC);
  float4* xr = (float4*)(x + (size_t)n * KC);
  float asum = 0.f;
#pragma unroll
  for (int g = 0; g < KC / 4; ++g) {
    float4 e = make_float4(0.f, 0.f, 0.f, 0.f);
    if (m) {
      float4 s = ysr[g];
      e.x = ((g * 4 + 0) == lab ? 1.f : 0.f) - s.x;
      e.y = ((g * 4 + 1) == lab ? 1.f : 0.f) - s.y;
      e.z = ((g * 4 + 2) == lab ? 1.f : 0.f) - s.z;
      e.w = ((g * 4 + 3) == lab ? 1.f : 0.f) - s.w;
      asum += fabsf(e.x) + fabsf(e.y) + fabsf(e.z) + fabsf(e.w);
    }
    xr[g] = e;
  }
  atomicAdd(&scal[0], asum);
  if (m) atomicAdd(&scal[1], 1.f);
}

// Core pull gather: acc = sum_{e in CSR[n]} norm_e * src[col_e, 4g..4g+3]
__device__ __forceinline__ float4 gather4(const int2 sg, const int2* __restrict__ epack,
                                          const float* __restrict__ src, int g) {
  float4 acc = make_float4(0.f, 0.f, 0.f, 0.f);
  const int s = sg.x, c = sg.y;
  for (int k = 0; k < c; ++k) {
    const int2 p = epack[s + k];                 // broadcast across the row's 10 lanes
    const float wv = __int_as_float(p.y);
    const float4 v = *(const float4*)(src + (size_t)p.x * KC + g * 4);
    acc.x = fmaf(wv, v.x, acc.x);
    acc.y = fmaf(wv, v.y, acc.y);
    acc.z = fmaf(wv, v.z, acc.z);
    acc.w = fmaf(wv, v.w, acc.w);
  }
  return acc;
}

// Correct layer, fully fused: dst = clip(alpha*acc + (1-alpha)*src, -1, 1)
__global__ void __launch_bounds__(256)
gather_clip_kernel(const int2* __restrict__ seg, const int2* __restrict__ epack,
                   const float* __restrict__ src, float* __restrict__ dst,
                   int N, float alpha) {
  const int tid = blockIdx.x * blockDim.x + threadIdx.x;
  if (tid >= N * (KC / 4)) return;
  const int n = tid / (KC / 4);
  const int g = tid % (KC / 4);
  const float4 acc = gather4(seg[n], epack, src, g);
  const float4 xv = *(const float4*)(src + (size_t)n * KC + g * 4);
  const float b = 1.f - alpha;
  float4 r;
  r.x = fminf(fmaxf(fmaf(alpha, acc.x, b * xv.x), -1.f), 1.f);
  r.y = fminf(fmaxf(fmaf(alpha, acc.y, b * xv.y), -1.f), 1.f);
  r.z = fminf(fmaxf(fmaf(alpha, acc.z, b * xv.z), -1.f), 1.f);
  r.w = fminf(fmaxf(fmaf(alpha, acc.w, b * xv.w), -1.f), 1.f);
  *(float4*)(dst + (size_t)n * KC + g * 4) = r;
}

// Smooth layer, fully fused: dst = softmax(alpha*acc + (1-alpha)*src) rowwise.
// blockDim = 320 = 10 waves: 32 nodes/block, 10 threads (float4-groups)/node;
// the row is exchanged through a 5KB LDS tile, one barrier.
__global__ void __launch_bounds__(320)
smooth_fused_kernel(const int2* __restrict__ seg, const int2* __restrict__ epack,
                    const float* __restrict__ src, float* __restrict__ dst,
                    int N, float alpha) {
  __shared__ float s_t[32][KC];  // 32 rows x 40 floats (row stride 160B, 16B aligned)
  const int t  = threadIdx.x;
  const int nl = t / (KC / 4);   // local node 0..31
  const int g  = t % (KC / 4);
  const int n  = blockIdx.x * 32 + nl;

  float4 tv = make_float4(0.f, 0.f, 0.f, 0.f);
  if (n < N) {
    const float4 acc = gather4(seg[n], epack, src, g);
    const float4 xv = *(const float4*)(src + (size_t)n * KC + g * 4);
    const float b = 1.f - alpha;
    tv.x = fmaf(alpha, acc.x, b * xv.x);
    tv.y = fmaf(alpha, acc.y, b * xv.y);
    tv.z = fmaf(alpha, acc.z, b * xv.z);
    tv.w = fmaf(alpha, acc.w, b * xv.w);
    *(float4*)(&s_t[nl][g * 4]) = tv;
  }
  __syncthreads();
  if (n < N) {
    float row[KC];
#pragma unroll
    for (int gg = 0; gg < KC / 4; ++gg) {
      const float4 v = *(const float4*)(&s_t[nl][gg * 4]);
      row[gg * 4 + 0] = v.x; row[gg * 4 + 1] = v.y;
      row[gg * 4 + 2] = v.z; row[gg * 4 + 3] = v.w;
    }
    float mx = -3.402823466e38f;
#pragma unroll
    for (int c = 0; c < KC; ++c) mx = fmaxf(mx, row[c]);
    float sum = 0.f;
#pragma unroll
    for (int c = 0; c < KC; ++c) sum += expf(row[c] - mx);
    const float inv = 1.f / sum;
    float4 o;
    o.x = expf(tv.x - mx) * inv;
    o.y = expf(tv.y - mx) * inv;
    o.z = expf(tv.z - mx) * inv;
    o.w = expf(tv.w - mx) * inv;
    *(float4*)(dst + (size_t)n * KC + g * 4) = o;
  }
}

// Per-node: denom=sum|smoothed|; scale; y_s = mask ? one_hot : y_soft + scale*smoothed
__global__ void apply_scale_kernel(const int* __restrict__ yt, const float* __restrict__ ys,
                                   const unsigned char* __restrict__ mask,
                                   float* __restrict__ x, const float* __restrict__ scal, int N) {
  int n = blockIdx.x * blockDim.x + threadIdx.x;
  if (n >= N) return;
  const float sigma = scal[0] / scal[1];
  float4* xr = (float4*)(x + (size_t)n * KC);
  const float4* ysr = (const float4*)(ys + (size_t)n * KC);
  float4 sm[KC / 4];
  float denom = 0.f;
#pragma unroll
  for (int g = 0; g < KC / 4; ++g) {
    float4 v = xr[g];
    sm[g] = v;
    denom += fabsf(v.x) + fabsf(v.y) + fabsf(v.z) + fabsf(v.w);
  }
  const float raw   = sigma / (denom > 0.f ? denom : 1.f);
  const float scale = ((denom <= 0.f) || (raw > 1000.f)) ? 1.f : raw;
  const bool m  = mask[n] != 0;
  const int lab = yt[n];
#pragma unroll
  for (int g = 0; g < KC / 4; ++g) {
    float4 o;
    if (m) {
      o.x = (g * 4 + 0) == lab ? 1.f : 0.f;
      o.y = (g * 4 + 1) == lab ? 1.f : 0.f;
      o.z = (g * 4 + 2) == lab ? 1.f : 0.f;
      o.w = (g * 4 + 3) == lab ? 1.f : 0.f;
    } else {
      float4 s = ysr[g], v = sm[g];
      o.x = s.x + scale * v.x;
      o.y = s.y + scale * v.y;
      o.z = s.z + scale * v.z;
      o.w = s.w + scale * v.w;
    }
    xr[g] = o;
  }
}

// ---------------------------------------------------------------------------
extern "C" void kernel_launch(void* const* d_in, const int* in_sizes, int n_in,
                              void* d_out, int out_size, void* d_ws, size_t ws_size,
                              hipStream_t stream) {
  const int*           y_true = (const int*)d_in[0];
  const float*         y_soft = (const float*)d_in[1];
  const unsigned char* mask   = (const unsigned char*)d_in[2];  // jnp bool_ = 1 byte
  const int*           eidx   = (const int*)d_in[3];
  const float*         ew     = (const float*)d_in[4];

  const int N = in_sizes[0];
  const int E = in_sizes[4];
  const int* row = eidx;      // edge_index[0]
  const int* col = eidx + E;  // edge_index[1]

  // --- carve workspace (256B aligned regions) ---
  char* ws = (char*)d_ws;
  size_t off = 0;
  auto carve = [&](size_t bytes) -> void* {
    void* p = ws + off;
    off = (off + bytes + 255) & ~(size_t)255;
    return p;
  };
  int2*  epack = (int2*)carve((size_t)E * 8);       // packed (col, norm)
  float* deg   = (float*)carve((size_t)N * 4);      // becomes dis after rsqrt
  int*   count = (int*)carve((size_t)N * 4);        // in-degree
  int2*  seg   = (int2*)carve((size_t)N * 8);       // (start, count)
  int*   wcur  = (int*)carve((size_t)N * 4);        // fill cursors
  float* scal  = (float*)carve(64);                 // [0]=sum|err| [1]=numel [2]=cursor
  float* xA    = (float*)carve((size_t)N * KC * 4);
  float* xB    = (float*)carve((size_t)N * KC * 4);

  const int T = 256;
  const int nNodeBlk   = (N + T - 1) / T;
  const int nEdgeBlk   = (E + T - 1) / T;
  const int nWork      = N * (KC / 4);              // (node, float4-group) threads
  const int nWorkBlk   = (nWork + T - 1) / T;
  const int nSmoothBlk = (N + 31) / 32;             // 32 nodes per 320-thread block

  // --- CSR build + GCN normalization (one-time, ~0.2% of a layer) ---
  zero4_kernel<<<(N / 4 + T - 1) / T, T, 0, stream>>>((float4*)deg, N / 4);
  zero4_kernel<<<(N / 4 + T - 1) / T, T, 0, stream>>>((float4*)count, N / 4);
  zero4_kernel<<<1, T, 0, stream>>>((float4*)scal, 4);
  degcnt_kernel<<<nEdgeBlk, T, 0, stream>>>(row, ew, deg, count, E);
  rsqrt_kernel<<<nNodeBlk, T, 0, stream>>>(deg, N);
  alloc_kernel<<<nNodeBlk, T, 0, stream>>>(count, seg, wcur, (int*)scal + 2, N);
  fill_kernel<<<nEdgeBlk, T, 0, stream>>>(row, col, ew, deg, wcur, epack, E);

  // --- error + sigma stats ---
  init_error_kernel<<<nNodeBlk, T, 0, stream>>>(y_true, y_soft, mask, xA, scal, N);

  float* cur = xA;
  float* alt = xB;

  // --- correct phase: 10 layers, alpha=0.9, clip; fused, atomic-free ---
  for (int it = 0; it < 10; ++it) {
    gather_clip_kernel<<<nWorkBlk, T, 0, stream>>>(seg, epack, cur, alt, N, 0.9f);
    float* tmp = cur; cur = alt; alt = tmp;
  }

  // --- scale + base signal y_s (in place into cur) ---
  apply_scale_kernel<<<nNodeBlk, T, 0, stream>>>(y_true, y_soft, mask, cur, scal, N);

  // --- smooth phase: 10 layers, alpha=0.8, fused softmax; last -> d_out ---
  for (int it = 0; it < 10; ++it) {
    float* dst = (it == 9) ? (float*)d_out : alt;
    smooth_fused_kernel<<<nSmoothBlk, 320, 0, stream>>>(seg, epack, cur, dst, N, 0.8f);
    if (it < 9) { float* tmp = cur; cur = alt; alt = tmp; }
  }
}